// AttentionGCN_18107582119990
// MI455X (gfx1250) — compile-verified
//
#include <hip/hip_runtime.h>

#define N_NODES 100000
#define N_EDGES 1600000

typedef __attribute__((ext_vector_type(2))) float v2f;
typedef __attribute__((ext_vector_type(8))) float v8f;

// ---------------------------------------------------------------- degree ----
__global__ void k_init_deg(float* __restrict__ deg) {
  int i = blockIdx.x * blockDim.x + threadIdx.x;
  if (i < N_NODES) deg[i] = 1.0f;              // self-loop baseline
}

__global__ void k_count_deg(const long long* __restrict__ ei,
                            float* __restrict__ deg) {
  int e = blockIdx.x * blockDim.x + threadIdx.x;
  if (e < N_EDGES) {
    int d = (int)ei[N_EDGES + e];              // dst row of edge_index
    atomicAdd(&deg[d], 1.0f);
  }
}

// dinv = rsqrt(deg) (in place), agg_x initialized with self-loop term d^-1 * x
__global__ void k_dinv_self(const float* __restrict__ x,
                            float* __restrict__ dinv,
                            float* __restrict__ aggx) {
  int i = blockIdx.x * blockDim.x + threadIdx.x;
  if (i < N_NODES) {
    float dv = rsqrtf(dinv[i]);
    dinv[i] = dv;
    float s = dv * dv;
    const float* xr = x + (long)i * 5;
    float* ar = aggx + (long)i * 5;
#pragma unroll
    for (int k = 0; k < 5; ++k) ar[k] = s * xr[k];
  }
}

// ------------------------------------------------- 5-wide edge aggregation --
__global__ void k_scatter_x(const long long* __restrict__ ei,
                            const float* __restrict__ x,
                            const float* __restrict__ dinv,
                            float* __restrict__ aggx) {
  int e = blockIdx.x * blockDim.x + threadIdx.x;
  if (e < N_EDGES) {
    int s = (int)ei[e];
    int d = (int)ei[N_EDGES + e];
    float nrm = dinv[s] * dinv[d];
    const float* xs = x + (long)s * 5;
    float* od = aggx + (long)d * 5;
#pragma unroll
    for (int k = 0; k < 5; ++k) atomicAdd(&od[k], nrm * xs[k]);
  }
}

// ----------------- fused MLP: h1 = relu(aggx@W1 + b1); p = h1@W2 (WMMA f32) --
// One wave32 per 16-node tile. A = 16x4 f32 (v2f/lane), B = 4x16 f32 (v2f/lane),
// C/D = 16x16 f32 (v8f/lane) per ISA 7.12.2 layouts. K=5 padded to two K=4 steps.
__global__ __launch_bounds__(256)
void k_mlp_wmma(const float* __restrict__ aggx, const float* __restrict__ W1,
                const float* __restrict__ b1v, const float* __restrict__ W2,
                const float* __restrict__ dinv,
                float* __restrict__ p, float* __restrict__ aggp) {
  __shared__ float lds[8][64][17];                 // [wave][col n][node], pad 17
  const int lane = threadIdx.x & 31;
  const int w    = threadIdx.x >> 5;
  const int half = lane >> 4;                      // 0: lanes 0-15, 1: 16-31
  const int mr   = lane & 15;
  const int m0   = (blockIdx.x * 8 + w) * 16;      // tile base node

  // A tile load (clamped index keeps EXEC all-1s; N % 16 == 0 so no straddle)
  int m = m0 + mr; if (m > N_NODES - 1) m = N_NODES - 1;
  const float* row = aggx + (long)m * 5;
  v2f a0, a1;
  if (half == 0) { a0[0] = row[0]; a0[1] = row[1]; a1[0] = row[4]; a1[1] = 0.f; }
  else           { a0[0] = row[2]; a0[1] = row[3]; a1[0] = 0.f;    a1[1] = 0.f; }

  float (*hw)[17] = lds[w];
#pragma unroll
  for (int t = 0; t < 4; ++t) {                    // four 16-col tiles of W1
    int n = t * 16 + mr;
    v2f bk0, bk1;
    if (half == 0) { bk0[0] = W1[0 * 64 + n]; bk0[1] = W1[1 * 64 + n];
                     bk1[0] = W1[4 * 64 + n]; bk1[1] = 0.f; }
    else           { bk0[0] = W1[2 * 64 + n]; bk0[1] = W1[3 * 64 + n];
                     bk1[0] = 0.f;            bk1[1] = 0.f; }
    v8f c = {};
    c = __builtin_amdgcn_wmma_f32_16x16x4_f32(false, a0, false, bk0,
                                              (short)0, c, false, false);
    c = __builtin_amdgcn_wmma_f32_16x16x4_f32(false, a1, false, bk1,
                                              (short)0, c, false, false);
    float bias = b1v[n];
#pragma unroll
    for (int r = 0; r < 8; ++r) {                  // row M = r + half*8, col n
      float v = c[r] + bias;
      v = v > 0.f ? v : 0.f;                       // relu
      hw[n][r + half * 8] = v;
    }
  }
  __syncthreads();

  // 32 lanes = 16 nodes x 2 classes: p[m][cls] = sum_n h1[m][n] * W2[n][cls]
  const int node = mr, cls = half;
  float acc = 0.f;
#pragma unroll 8
  for (int n = 0; n < 64; ++n) acc += hw[n][node] * W2[n * 2 + cls];
  int mm = m0 + node;
  if (mm < N_NODES) {
    float dv = dinv[mm];
    p[(long)mm * 2 + cls]    = acc;
    aggp[(long)mm * 2 + cls] = dv * dv * acc;      // conv2 self-loop term
  }
}

// ------------------------------------------------- 2-wide edge aggregation --
__global__ void k_scatter_p(const long long* __restrict__ ei,
                            const float* __restrict__ p,
                            const float* __restrict__ dinv,
                            float* __restrict__ aggp) {
  int e = blockIdx.x * blockDim.x + threadIdx.x;
  if (e < N_EDGES) {
    int s = (int)ei[e];
    int d = (int)ei[N_EDGES + e];
    float nrm = dinv[s] * dinv[d];
    atomicAdd(&aggp[(long)d * 2 + 0], nrm * p[(long)s * 2 + 0]);
    atomicAdd(&aggp[(long)d * 2 + 1], nrm * p[(long)s * 2 + 1]);
  }
}

// ------------------------------------------------------------ log_softmax --
__global__ void k_logsoftmax(const float* __restrict__ aggp,
                             const float* __restrict__ b2,
                             float* __restrict__ out) {
  int i = blockIdx.x * blockDim.x + threadIdx.x;
  if (i < N_NODES) {
    float z0 = aggp[(long)i * 2 + 0] + b2[0];
    float z1 = aggp[(long)i * 2 + 1] + b2[1];
    float mx = fmaxf(z0, z1);
    float lse = mx + logf(expf(z0 - mx) + expf(z1 - mx));
    out[(long)i * 2 + 0] = z0 - lse;
    out[(long)i * 2 + 1] = z1 - lse;
  }
}

// ------------------------------------------------------------------ launch --
extern "C" void kernel_launch(void* const* d_in, const int* in_sizes, int n_in,
                              void* d_out, int out_size, void* d_ws, size_t ws_size,
                              hipStream_t stream) {
  (void)in_sizes; (void)n_in; (void)out_size; (void)ws_size;
  const float*     x  = (const float*)d_in[0];
  const long long* ei = (const long long*)d_in[1];
  // d_in[2] (aff_w) / d_in[3] (cog_w): softmax over a single column == 1 -> dead
  const float* W1 = (const float*)d_in[4];
  const float* b1 = (const float*)d_in[5];
  const float* W2 = (const float*)d_in[6];
  const float* b2 = (const float*)d_in[7];
  float* out = (float*)d_out;

  // workspace layout (floats, 256B-aligned regions): ~4.0 MB total
  float* ws   = (float*)d_ws;
  float* dinv = ws;                    // 100000 (deg, then rsqrt(deg) in place)
  float* aggx = dinv + 100096;         // 500000
  float* p    = aggx + 500096;         // 200000
  float* aggp = p    + 200064;         // 200000

  const int TB = 256;
  k_init_deg   <<<(N_NODES + TB - 1) / TB, TB, 0, stream>>>(dinv);
  k_count_deg  <<<(N_EDGES + TB - 1) / TB, TB, 0, stream>>>(ei, dinv);
  k_dinv_self  <<<(N_NODES + TB - 1) / TB, TB, 0, stream>>>(x, dinv, aggx);
  k_scatter_x  <<<(N_EDGES + TB - 1) / TB, TB, 0, stream>>>(ei, x, dinv, aggx);
  k_mlp_wmma   <<<(N_NODES + 127) / 128,   TB, 0, stream>>>(aggx, W1, b1, W2,
                                                            dinv, p, aggp);
  k_scatter_p  <<<(N_EDGES + TB - 1) / TB, TB, 0, stream>>>(ei, p, dinv, aggp);
  k_logsoftmax <<<(N_NODES + TB - 1) / TB, TB, 0, stream>>>(aggp, b2, out);
}